// Criterion_67954972557709
// MI455X (gfx1250) — compile-verified
//
#include <hip/hip_runtime.h>

// ---------------------------------------------------------------------------
// dist(i,j) = |xi|^2 + |xj|^2 - 2*<xi,xj> + 1e-8 ; sum of -log(dist) where
// dist <= 0.25, i != j, (i,j) not an edge (either direction).
// Cross term <xi,xj> computed 16x16 tiles at a time with V_WMMA_F32_16X16X4_F32
// (K=3 padded to 4). Edge exclusion handled as a deduplicated subtraction so
// the N^2 hot loop stays mask-free. Log via raw v_log_f32; epilogue fully
// branchless (all masking via v_cndmask, no exec-mask regions).
// ---------------------------------------------------------------------------

typedef __attribute__((ext_vector_type(2))) float v2f;
typedef __attribute__((ext_vector_type(8))) float v8f;

#define NPTS     8192
#define NTILES   512          // NPTS / 16
#define THRESH2  0.25f
#define DIST_EPS 1e-8f
#define NEG_LN2  (-0.69314718055994530942f)

#define WS_NORM_OFF 0         // 8192 f32  = 32 KB
#define WS_ACC_OFF  32768     // 1 double
#define WS_TAB_OFF  40960     // hash table
#define TAB_ENTRIES 262144u   // u32 keys, power of two, 1 MB

// -log(d) with d guaranteed >= 1e-8 (no denorm/inf fixup needed):
// v_log_f32 computes log2, scale by ln2.
__device__ __forceinline__ float neg_log_clamped(float d) {
    d = fmaxf(d, DIST_EPS);
    return NEG_LN2 * __builtin_amdgcn_logf(d);
}

// ---- squared norms ---------------------------------------------------------
__global__ void norms_kernel(const float* __restrict__ pos,
                             float* __restrict__ nrm) {
    int i = blockIdx.x * blockDim.x + threadIdx.x;
    if (i < NPTS) {
        float x = pos[3 * i + 0], y = pos[3 * i + 1], z = pos[3 * i + 2];
        nrm[i] = fmaf(x, x, fmaf(y, y, z * z));
    }
}

// ---- main N^2 pass: WMMA tiles of X * X^T ---------------------------------
__global__ __launch_bounds__(256)
void pair_kernel(const float* __restrict__ pos,
                 const float* __restrict__ nrm,
                 double* __restrict__ acc) {
    const int lane = threadIdx.x & 31;   // wave32
    const int wave = threadIdx.x >> 5;   // 8 waves / block
    const int half = lane >> 4;          // 0: lanes 0-15, 1: lanes 16-31
    const int l16  = lane & 15;

    // A matrix (16x4 f32): lanes 0-15 hold (x,y) of row M=l16 in v0/v1,
    // lanes 16-31 hold (z, 0).  B (4x16) mirrors this per-column.
    const int itile = blockIdx.x;
    const int rowA  = itile * 16 + l16;
    const float ax = pos[3 * rowA + 0];   // unconditional b96 load, then select
    const float ay = pos[3 * rowA + 1];
    const float az = pos[3 * rowA + 2];
    v2f A;
    A.x = half ? az : ax;
    A.y = half ? 0.0f : ay;

    // norms + global row ids of the 8 M-rows this lane's accumulators map to
    const int gibase = itile * 16 + 8 * half;
    float ni[8];
#pragma unroll
    for (int r = 0; r < 8; ++r)
        ni[r] = nrm[gibase + r];

    float accf = 0.0f;
    for (int jt = blockIdx.y * 8 + wave; jt < NTILES; jt += gridDim.y * 8) {
        const int colB = jt * 16 + l16;
        const float bx = pos[3 * colB + 0];
        const float by = pos[3 * colB + 1];
        const float bz = pos[3 * colB + 2];
        v2f B;
        B.x = half ? bz : bx;
        B.y = half ? 0.0f : by;
        const float nj = nrm[colB];            // N = lane&15 -> col id = colB

        v8f c = {};
        // D = A x B : 16x16 tile of dot products
        c = __builtin_amdgcn_wmma_f32_16x16x4_f32(
                /*neg_a=*/false, A, /*neg_b=*/false, B,
                /*c_mod=*/(short)0, c, /*reuse_a=*/false, /*reuse_b=*/false);

        const int gj = colB;                   // global column id for this lane
#pragma unroll
        for (int r = 0; r < 8; ++r) {
            float dist = fmaf(-2.0f, c[r], ni[r] + nj) + DIST_EPS;
            float ml = neg_log_clamped(dist);
            bool keep = (dist <= THRESH2) && (gibase + r != gj);
            accf += keep ? ml : 0.0f;          // pure v_cndmask, no branches
        }
    }

    // wave32 reduction, then block reduction, one f64 atomic per block
    for (int off = 16; off > 0; off >>= 1)
        accf += __shfl_down(accf, off, 32);
    __shared__ float wsum[8];
    if (lane == 0) wsum[wave] = accf;
    __syncthreads();
    if (threadIdx.x == 0) {
        float s = 0.0f;
#pragma unroll
        for (int w = 0; w < 8; ++w) s += wsum[w];
        atomicAdd(acc, (double)s);
    }
}

// ---- edge correction: dedup unordered pairs, subtract their contribution --
__global__ void edge_kernel(const float* __restrict__ pos,
                            const int* __restrict__ edges,
                            const float* __restrict__ nrm,
                            unsigned* __restrict__ tab,
                            double* __restrict__ acc, int E) {
    int e = blockIdx.x * blockDim.x + threadIdx.x;
    if (e >= E) return;
    int a = edges[2 * e + 0];
    int b = edges[2 * e + 1];
    if (a == b) return;                       // diagonal already excluded
    int lo = a < b ? a : b;
    int hi = a < b ? b : a;
    unsigned key = (unsigned)lo * (unsigned)NPTS + (unsigned)hi + 1u;
    unsigned h = (key * 2654435761u) & (TAB_ENTRIES - 1u);
    bool owner = false;
    for (;;) {
        unsigned prev = atomicCAS(&tab[h], 0u, key);
        if (prev == 0u)  { owner = true; break; }
        if (prev == key) { break; }           // duplicate edge / reversed dup
        h = (h + 1u) & (TAB_ENTRIES - 1u);
    }
    if (!owner) return;
    // same dist + log formula as the main pass so the subtraction cancels
    float dot = fmaf(pos[3 * a + 0], pos[3 * b + 0],
                fmaf(pos[3 * a + 1], pos[3 * b + 1],
                     pos[3 * a + 2] * pos[3 * b + 2]));
    float dist = fmaf(-2.0f, dot, nrm[a] + nrm[b]) + DIST_EPS;
    if (dist <= THRESH2) {
        float ml = neg_log_clamped(dist);
        // main pass added ml for (a,b) and (b,a): subtract both
        atomicAdd(acc, (double)(-2.0f * ml));
    }
}

// ---- finalize --------------------------------------------------------------
__global__ void finalize_kernel(const double* __restrict__ acc,
                                float* __restrict__ out) {
    out[0] = (float)(acc[0]);   // WEIGHT == 1.0
}

extern "C" void kernel_launch(void* const* d_in, const int* in_sizes, int n_in,
                              void* d_out, int out_size, void* d_ws, size_t ws_size,
                              hipStream_t stream) {
    const float* pos   = (const float*)d_in[0];
    const int*   edges = (const int*)d_in[1];
    const int    E     = in_sizes[1] / 2;

    char*     ws  = (char*)d_ws;
    float*    nrm = (float*)(ws + WS_NORM_OFF);
    double*   acc = (double*)(ws + WS_ACC_OFF);
    unsigned* tab = (unsigned*)(ws + WS_TAB_OFF);

    // zero accumulator + hash table (ws is poisoned; must re-zero every call)
    hipMemsetAsync(ws + WS_ACC_OFF, 0,
                   (WS_TAB_OFF - WS_ACC_OFF) + TAB_ENTRIES * sizeof(unsigned),
                   stream);

    norms_kernel<<<NPTS / 256, 256, 0, stream>>>(pos, nrm);
    edge_kernel<<<(E + 255) / 256, 256, 0, stream>>>(pos, edges, nrm, tab, acc, E);
    pair_kernel<<<dim3(NTILES, 16), 256, 0, stream>>>(pos, nrm, acc);
    finalize_kernel<<<1, 1, 0, stream>>>(acc, (float*)d_out);
}